// GINLayer_29025388986626
// MI455X (gfx1250) — compile-verified
//
#include <hip/hip_runtime.h>
#include <stdint.h>

#define NODES   50000
#define EDGES   800000
#define DIM     128
#define NEGINIT (-1e9f)
#define SLOPE   0.01f
#define WAVES   4
#define LSTRIDE 132   // 128 + 4 pad: conflict-free 16-row column reads (64 banks)

typedef float v2f __attribute__((ext_vector_type(2)));
typedef float v8f __attribute__((ext_vector_type(8)));

// Order-preserving float -> u32 map: max over u32 == max over f32
__device__ __forceinline__ unsigned enc_f32(float f) {
  unsigned u = __float_as_uint(f);
  return (u & 0x80000000u) ? ~u : (u | 0x80000000u);
}
__device__ __forceinline__ float dec_f32(unsigned u) {
  unsigned b = (u & 0x80000000u) ? (u ^ 0x80000000u) : ~u;
  return __uint_as_float(b);
}

__global__ void init_keys(unsigned* __restrict__ keys, int n4) {
  int i = blockIdx.x * blockDim.x + threadIdx.x;
  if (i < n4) {
    unsigned v = enc_f32(NEGINIT);
    ((uint4*)keys)[i] = make_uint4(v, v, v, v);
  }
}

// 32 lanes per edge; each lane covers 4 contiguous dims (float4 gather)
__global__ void edge_scatter(const float* __restrict__ x,
                             const long long* __restrict__ ei,
                             const float* __restrict__ ew,
                             unsigned* __restrict__ keys) {
  long long t = (long long)blockIdx.x * blockDim.x + threadIdx.x;
  int e = (int)(t >> 5);
  if (e >= EDGES) return;
  int q   = (int)(t & 31);
  int src = (int)ei[e];
  int dst = (int)ei[EDGES + e];
  float w = ew[e];
  const float4 xv = ((const float4*)(x + (size_t)src * DIM))[q];
  unsigned* kp = keys + (size_t)dst * DIM + q * 4;
  atomicMax(kp + 0, enc_f32(xv.x * w));
  atomicMax(kp + 1, enc_f32(xv.y * w));
  atomicMax(kp + 2, enc_f32(xv.z * w));
  atomicMax(kp + 3, enc_f32(xv.w * w));
}

// Fused: h = (1+eps)x + agg  ->  GEMM(w1)+b1 -> LeakyReLU -> GEMM(w2)+b2
// One wave owns 16 node rows; full fp32 WMMA 16x16x4.
__global__ __launch_bounds__(WAVES * 32) void gin_mlp(
    const float*    __restrict__ x,
    const unsigned* __restrict__ keys,
    const float*    __restrict__ w1, const float* __restrict__ b1,
    const float*    __restrict__ w2, const float* __restrict__ b2,
    const float*    __restrict__ epsp,
    float*          __restrict__ out) {
  __shared__ float lds[WAVES][16][LSTRIDE];

  const int wv   = threadIdx.x >> 5;
  const int lane = threadIdx.x & 31;
  const int rowbase = blockIdx.x * (WAVES * 16) + wv * 16;
  const float eps1 = 1.0f + epsp[0];

  // ---- Phase 1: build h tile (16 x 128) in LDS ----
  const int d0 = lane * 4;
  for (int r = 0; r < 16; ++r) {
    int row = rowbase + r;
    row = (row < NODES) ? row : (NODES - 1);           // clamp, no divergence
    const float4 xv = *(const float4*)(x + (size_t)row * DIM + d0);
    const uint4  kv = *(const uint4*)(keys + (size_t)row * DIM + d0);
    float a0 = dec_f32(kv.x), a1 = dec_f32(kv.y);
    float a2 = dec_f32(kv.z), a3 = dec_f32(kv.w);
    a0 = (a0 == NEGINIT) ? 0.f : a0;
    a1 = (a1 == NEGINIT) ? 0.f : a1;
    a2 = (a2 == NEGINIT) ? 0.f : a2;
    a3 = (a3 == NEGINIT) ? 0.f : a3;
    float4 h;
    h.x = eps1 * xv.x + a0;  h.y = eps1 * xv.y + a1;
    h.z = eps1 * xv.z + a2;  h.w = eps1 * xv.w + a3;
    *(float4*)(&lds[wv][r][d0]) = h;
  }
  __syncthreads();

  // Fragment coordinates (f32 WMMA 16x16x4 layouts)
  const int mrow = lane & 15;             // A: M index
  const int kb   = (lane >> 4) << 1;      // A/B: K sub-offset 0 or 2
  const int col  = lane & 15;             // B/C/D: N index
  const int moff = (lane >> 4) << 3;      // C/D: M base 0 or 8

  // ---- Phase 2: layer 1, all 8 N-tiles held in registers ----
  v8f acc[8];
  for (int nt = 0; nt < 8; ++nt) {
    v8f c = {};
    const float* wcol = w1 + (size_t)(nt * 16 + col) * DIM;
    for (int kk = 0; kk < 32; ++kk) {
      const int k = kk * 4 + kb;
      v2f a; a.x = lds[wv][mrow][k];  a.y = lds[wv][mrow][k + 1];
      v2f b; b.x = wcol[k];           b.y = wcol[k + 1];
      c = __builtin_amdgcn_wmma_f32_16x16x4_f32(false, a, false, b,
                                                (short)0, c, false, false);
    }
    acc[nt] = c;
  }
  __syncthreads();

  // bias + LeakyReLU, write h2 back into the same LDS tile
  for (int nt = 0; nt < 8; ++nt) {
    const float bb = b1[nt * 16 + col];
    for (int r = 0; r < 8; ++r) {
      float v = acc[nt][r] + bb;
      v = (v >= 0.f) ? v : SLOPE * v;
      lds[wv][moff + r][nt * 16 + col] = v;
    }
  }
  __syncthreads();

  // ---- Phase 3: layer 2 + bias + guarded store ----
  for (int nt = 0; nt < 8; ++nt) {
    v8f c = {};
    const float* wcol = w2 + (size_t)(nt * 16 + col) * DIM;
    for (int kk = 0; kk < 32; ++kk) {
      const int k = kk * 4 + kb;
      v2f a; a.x = lds[wv][mrow][k];  a.y = lds[wv][mrow][k + 1];
      v2f b; b.x = wcol[k];           b.y = wcol[k + 1];
      c = __builtin_amdgcn_wmma_f32_16x16x4_f32(false, a, false, b,
                                                (short)0, c, false, false);
    }
    const float bb = b2[nt * 16 + col];
    for (int r = 0; r < 8; ++r) {
      const int row = rowbase + moff + r;
      if (row < NODES)
        out[(size_t)row * DIM + nt * 16 + col] = c[r] + bb;
    }
  }
}

extern "C" void kernel_launch(void* const* d_in, const int* in_sizes, int n_in,
                              void* d_out, int out_size, void* d_ws, size_t ws_size,
                              hipStream_t stream) {
  const float*     x   = (const float*)d_in[0];
  const long long* ei  = (const long long*)d_in[1];
  const float*     ew  = (const float*)d_in[2];
  const float*     w1  = (const float*)d_in[3];
  const float*     b1  = (const float*)d_in[4];
  const float*     w2  = (const float*)d_in[5];
  const float*     b2  = (const float*)d_in[6];
  const float*     eps = (const float*)d_in[7];
  float*    out  = (float*)d_out;
  unsigned* keys = (unsigned*)d_ws;   // NODES*DIM u32 = 25.6 MB

  const int n4 = (NODES * DIM) / 4;                          // uint4 fills
  init_keys<<<(n4 + 255) / 256, 256, 0, stream>>>(keys, n4);

  const long long tot = (long long)EDGES * 32;               // 32 lanes/edge
  edge_scatter<<<(unsigned)((tot + 255) / 256), 256, 0, stream>>>(x, ei, ew, keys);

  gin_mlp<<<(NODES + WAVES * 16 - 1) / (WAVES * 16), WAVES * 32, 0, stream>>>(
      x, keys, w1, b1, w2, b2, eps, out);
}